// Model_68186900792186
// MI455X (gfx1250) — compile-verified
//
#include <hip/hip_runtime.h>
#include <stdint.h>

// Splice kernel for MI455X (gfx1250): out[m] = concat(a[m/4][:sa], b[m][:sb], dst[m][sa+sb:])
// Pure bandwidth problem (~270 MB moved -> ~12us at 23.3 TB/s roofline).
// Bulk data moves via CDNA5 async global->LDS->global B128 copies (ASYNCcnt),
// 4 chunks in flight per wait to maximize memory-level parallelism.

#define THREADS 256
#define VEC     4              // ints per chunk (16 B)
#define CHUNKS  4              // chunks per thread
#define BLOCK_ITEMS (THREADS * VEC * CHUNKS)   // 4096 ints per block

typedef int v4i __attribute__((ext_vector_type(4)));

// 32-bit LDS byte offset of a pointer to a __shared__ object.
__device__ __forceinline__ uint32_t lds_offset_of(void* p) {
    return (uint32_t)(uintptr_t)(__attribute__((address_space(3))) void*)p;
}

// Async copy: global -> LDS, 16 bytes per active lane (regular temporal; keeps
// the 4x-reused A table resident in L2).
__device__ __forceinline__ void async_gload_b128(const void* gaddr, uint32_t ldsoff) {
    asm volatile("global_load_async_to_lds_b128 %0, %1, off"
                 :
                 : "v"(ldsoff), "v"((unsigned long long)(uintptr_t)gaddr)
                 : "memory");
}

// Async copy: global -> LDS, non-temporal (read-once dst tail; don't evict A).
__device__ __forceinline__ void async_gload_b128_nt(const void* gaddr, uint32_t ldsoff) {
    asm volatile("global_load_async_to_lds_b128 %0, %1, off th:TH_LOAD_NT"
                 :
                 : "v"(ldsoff), "v"((unsigned long long)(uintptr_t)gaddr)
                 : "memory");
}

// Async copy: LDS -> global, 16 bytes per active lane.
__device__ __forceinline__ void async_gstore_b128(void* gaddr, uint32_t ldsoff) {
    asm volatile("global_store_async_from_lds_b128 %0, %1, off"
                 :
                 : "v"((unsigned long long)(uintptr_t)gaddr), "v"(ldsoff)
                 : "memory");
}

__device__ __forceinline__ void wait_asynccnt0() {
    asm volatile("s_wait_asynccnt 0x0" ::: "memory");
}

__global__ __launch_bounds__(THREADS)
void splice_rows_kernel(const int* __restrict__ dst,
                        const int* __restrict__ a,
                        const int* __restrict__ b,
                        const int* __restrict__ sla,
                        const int* __restrict__ slb,
                        int* __restrict__ out,
                        int LEN_A, int LEN_B, int LEN_OUT, int repShift)
{
    __shared__ int sbuf[BLOCK_ITEMS];   // 16 KB: one 16B slot per (chunk, lane)

    const int row = blockIdx.y;
    const int tid = threadIdx.x;

    // Block-uniform scalars (row uniform -> scalar loads; shift, no division).
    const int sa    = sla[row >> repShift];
    const int sb    = slb[row];
    const int total = sa + sb;

    const int* aRow = a   + (size_t)(row >> repShift) * LEN_A;
    const int* bRow = b   + (size_t)row * LEN_B;
    const int* dRow = dst + (size_t)row * LEN_OUT;
    int*       oRow = out + (size_t)row * LEN_OUT;

    const int base = blockIdx.x * BLOCK_ITEMS;

    int  jof[CHUNKS];       // chunk start column
    bool fast[CHUNKS];      // chunk staged through LDS?

    // Pass 1: issue all async loads; resolve boundary/B-region chunks directly.
#pragma unroll
    for (int c = 0; c < CHUNKS; ++c) {
        const int j0 = base + (c * THREADS + tid) * VEC;
        jof[c]  = j0;
        fast[c] = false;
        if (j0 >= LEN_OUT) continue;

        const uint32_t loff = lds_offset_of(&sbuf[(c * THREADS + tid) * VEC]);

        if (j0 + VEC <= sa) {
            // Entire chunk inside region A: 16B-aligned async stream.
            async_gload_b128(aRow + j0, loff);
            fast[c] = true;
        } else if (j0 >= total) {
            // Entire chunk inside pass-through tail: NT read (read-once).
            async_gload_b128_nt(dRow + j0, loff);
            fast[c] = true;
        } else {
            // Boundary chunk or B region (<= ~128 elements per row).
            v4i v;
#pragma unroll
            for (int k = 0; k < VEC; ++k) {
                const int j = j0 + k;
                int x;
                if (j < sa)         x = aRow[j];
                else if (j < total) x = bRow[j - sa];
                else                x = dRow[j];
                v[k] = x;
            }
            *(v4i*)(oRow + j0) = v;   // 16B-aligned vector store
        }
    }

    // Single wait for all in-flight async loads (up to 64 B/lane outstanding).
    wait_asynccnt0();

    // Pass 2: drain LDS to the output with async stores.
#pragma unroll
    for (int c = 0; c < CHUNKS; ++c) {
        if (fast[c]) {
            const uint32_t loff = lds_offset_of(&sbuf[(c * THREADS + tid) * VEC]);
            async_gstore_b128(oRow + jof[c], loff);
        }
    }
    // S_ENDPGM performs an implicit wait-idle, covering outstanding async stores.
}

extern "C" void kernel_launch(void* const* d_in, const int* in_sizes, int n_in,
                              void* d_out, int out_size, void* d_ws, size_t ws_size,
                              hipStream_t stream) {
    const int* dst = (const int*)d_in[0];   // (M, LEN_OUT)
    const int* a   = (const int*)d_in[1];   // (BS, LEN_A)
    const int* b   = (const int*)d_in[2];   // (M, LEN_B)
    const int* sla = (const int*)d_in[3];   // (BS,)
    const int* slb = (const int*)d_in[4];   // (M,)
    int*       out = (int*)d_out;           // (M, LEN_OUT) int32

    const int BS      = in_sizes[3];
    const int M       = in_sizes[4];
    const int LEN_A   = in_sizes[1] / BS;
    const int LEN_B   = in_sizes[2] / M;
    const int LEN_OUT = in_sizes[0] / M;
    const int rep     = M / BS;             // REPEAT_STEP (power of two)
    int repShift = 0;
    while ((1 << repShift) < rep) ++repShift;

    dim3 grid((LEN_OUT + BLOCK_ITEMS - 1) / BLOCK_ITEMS, M);
    splice_rows_kernel<<<grid, THREADS, 0, stream>>>(dst, a, b, sla, slb, out,
                                                     LEN_A, LEN_B, LEN_OUT, repShift);
}